// Rapidash_9259949490221
// MI455X (gfx1250) — compile-verified
//
#include <hip/hip_runtime.h>
#include <math.h>

#define DEV_INLINE __device__ __forceinline__

typedef __attribute__((ext_vector_type(2))) float v2f;
typedef __attribute__((ext_vector_type(8))) float v8f;

constexpr int N_   = 10000;
constexpr int E_   = 160000;
constexpr int O_   = 12;
constexpr int C_   = 64;
constexpr int KD_  = 64;
constexpr int KEY_ = 128;
constexpr int F_   = 256;          // WF*C
constexpr int NO_  = N_ * O_;      // 120000
constexpr int EO_  = E_ * O_;      // 1920000

// ---------------- CDNA5 FP32 WMMA 16x16x4 ----------------
// A (16x4 f32): lane L holds A[M=L%16][K = r + 2*(L/16)] in vgpr r
// B (4x16 f32): lane L holds B[K = r + 2*(L/16)][N=L%16] in vgpr r
// C/D (16x16):  lane L, vgpr v -> [M = v + 8*(L/16)][N = L%16]
DEV_INLINE v8f wmma4(v2f a, v2f b, v8f c) {
  return __builtin_amdgcn_wmma_f32_16x16x4_f32(false, a, false, b, (short)0, c,
                                               false, false);
}

DEV_INLINE v2f ld2(const float* p) {
  float2 t = *reinterpret_cast<const float2*>(p);
  v2f r; r.x = t.x; r.y = t.y; return r;
}

DEV_INLINE void atomicMaxF(float* addr, float val) {
  int old = __float_as_int(*addr);
  while (__int_as_float(old) < val) {
    int assumed = old;
    old = atomicCAS((int*)addr, assumed, __float_as_int(val));
    if (old == assumed) break;
  }
}

// ---------------- init: zero x1, denom=1e-6, gmax=-inf ----------------
__global__ void k_init(float* __restrict__ denom, float* __restrict__ x1,
                       float* __restrict__ gmax) {
  int i = blockIdx.x * blockDim.x + threadIdx.x;
  if (i < NO_ * C_) x1[i] = 0.0f;
  if (i < NO_)      denom[i] = 1e-6f;
  if (i == 0)       gmax[0] = -INFINITY;
}

// ---------------- fk[p,o,c] = sum_k fkb[p,o,k] * fiber_w[c,k] ----------------
__global__ void k_fk(const float* __restrict__ fkb,
                     const float* __restrict__ fiber_w, float* __restrict__ fk) {
  int i = blockIdx.x * blockDim.x + threadIdx.x;
  if (i >= O_ * O_ * C_) return;
  int po = i / C_, c = i % C_;
  float s = 0.f;
#pragma unroll 8
  for (int k = 0; k < KD_; ++k)
    s = fmaf(fkb[po * KD_ + k], fiber_w[c * KD_ + k], s);
  fk[i] = s;
}

// ---------------- keys/qrys: [NO,64] @ W^T[64,128] + bias ----------------
__global__ void k_qk(const float* __restrict__ x,
                     const float* __restrict__ key_w, const float* __restrict__ key_b,
                     const float* __restrict__ query_w, const float* __restrict__ query_b,
                     float* __restrict__ keys, float* __restrict__ qrys) {
  const int MT = NO_ / 16;   // 7500
  const int NT = KEY_ / 16;  // 8
  int gt = blockIdx.x * blockDim.x + threadIdx.x;
  int wave = gt >> 5, lane = gt & 31;
  if (wave >= MT * NT * 2) return;
  int which = wave / (MT * NT);
  int rem   = wave % (MT * NT);
  int mt = rem / NT, nt = rem % NT;
  const float* W  = which ? query_w : key_w;
  const float* Bb = which ? query_b : key_b;
  float*       Ot = which ? qrys    : keys;
  int m = lane & 15, half = lane >> 4;
  int row0 = mt * 16;
  int col  = nt * 16 + m;
  const float* ap = x + (size_t)(row0 + m) * C_ + half * 2;
  const float* bp = W + (size_t)col * C_ + half * 2;   // B[k][n] = W[n*64+k]
  v8f acc = {};
#pragma unroll
  for (int k0 = 0; k0 < C_; k0 += 4)
    acc = wmma4(ld2(ap + k0), ld2(bp + k0), acc);
  float bias = Bb[col];
#pragma unroll
  for (int v = 0; v < 8; ++v)
    Ot[(size_t)(row0 + v + 8 * half) * KEY_ + col] = acc[v] + bias;
}

// ---------------- logits[e,o] = <keys[src,o,:], qrys[dst,o,:]>/sqrt(KEY); global max ----------------
__global__ void k_logits(const float* __restrict__ keys, const float* __restrict__ qrys,
                         const int* __restrict__ src, const int* __restrict__ dst,
                         float* __restrict__ logits, float* __restrict__ gmax) {
  int i = blockIdx.x * blockDim.x + threadIdx.x;
  float lg = -INFINITY;
  if (i < EO_) {
    int e = i / O_, o = i % O_;
    const float4* kr = (const float4*)(keys + ((size_t)src[e] * O_ + o) * KEY_);
    const float4* qr = (const float4*)(qrys + ((size_t)dst[e] * O_ + o) * KEY_);
    float s = 0.f;
#pragma unroll
    for (int k = 0; k < KEY_ / 4; ++k) {
      float4 a = kr[k], b = qr[k];
      s += a.x * b.x + a.y * b.y + a.z * b.z + a.w * b.w;
    }
    lg = s * 0.08838834764831845f;  // 1/sqrt(128)
    logits[i] = lg;
  }
  __shared__ float red[256];
  red[threadIdx.x] = lg;
  __syncthreads();
  for (int s2 = 128; s2 > 0; s2 >>= 1) {
    if ((int)threadIdx.x < s2)
      red[threadIdx.x] = fmaxf(red[threadIdx.x], red[threadIdx.x + s2]);
    __syncthreads();
  }
  if (threadIdx.x == 0) atomicMaxF(gmax, red[0]);
}

// ---------------- ev = exp(logit - gmax); denom[dst,o] += ev ----------------
__global__ void k_expdenom(float* __restrict__ ev, const int* __restrict__ dst,
                           const float* __restrict__ gmax, float* __restrict__ denom) {
  int i = blockIdx.x * blockDim.x + threadIdx.x;
  if (i >= EO_) return;
  int e = i / O_, o = i % O_;
  float v = expf(ev[i] - gmax[0]);
  ev[i] = v;
  atomicAdd(&denom[dst[e] * O_ + o], v);
}

// ---------------- fused edge GEMM + attention weighting + scatter-add ----------------
// rows = (e,o) pairs of kernel_basis[E*O,64]; per wave: 16 rows x 64 cols (4 WMMA tiles)
__global__ void k_edge(const float* __restrict__ kb, const float* __restrict__ kernel_w,
                       const int* __restrict__ dst, const float* __restrict__ ev,
                       const float* __restrict__ denom, float* __restrict__ x1) {
  const int TILES = EO_ / 16;  // 120000
  int gt = blockIdx.x * blockDim.x + threadIdx.x;
  int wave = gt >> 5, lane = gt & 31;
  if (wave >= TILES) return;
  int m = lane & 15, half = lane >> 4;
  int row0 = wave * 16;
  const float* ap = kb + (size_t)(row0 + m) * KD_ + half * 2;
  v8f acc[4] = {};
#pragma unroll 4
  for (int k0 = 0; k0 < KD_; k0 += 4) {
    v2f a = ld2(ap + k0);
#pragma unroll
    for (int t = 0; t < 4; ++t) {
      int col = t * 16 + m;                           // B[k][c] = kernel_w[c*64+k]
      v2f b = ld2(kernel_w + (size_t)col * KD_ + half * 2 + k0);
      acc[t] = wmma4(a, b, acc[t]);
    }
  }
  float w[8]; int ob[8];
#pragma unroll
  for (int v = 0; v < 8; ++v) {
    int r = row0 + v + 8 * half;
    int e = r / O_, o = r - e * O_;
    int no = dst[e] * O_ + o;
    w[v]  = ev[r] / denom[no];
    ob[v] = no * C_;
  }
#pragma unroll
  for (int t = 0; t < 4; ++t) {
    int col = t * 16 + m;
#pragma unroll
    for (int v = 0; v < 8; ++v)
      atomicAdd(&x1[ob[v] + col], acc[t][v] * w[v]);
  }
}

// ---------------- fiber conv (depthwise over O) + LayerNorm, fused ----------------
__global__ void k_fiberln(const float* __restrict__ x1, const float* __restrict__ fk,
                          const float* __restrict__ conv_bias,
                          const float* __restrict__ gamma, const float* __restrict__ beta,
                          float* __restrict__ h) {
  int np = blockIdx.x;        // n*O + p
  int n = np / O_, p = np % O_;
  int c = threadIdx.x;        // 0..63
  float acc = 0.f;
#pragma unroll
  for (int o = 0; o < O_; ++o)
    acc = fmaf(x1[((size_t)n * O_ + o) * C_ + c], fk[(p * O_ + o) * C_ + c], acc);
  float val = acc * (1.0f / O_) + conv_bias[c];
  __shared__ float s1[64], s2[64];
  s1[c] = val; s2[c] = val * val;
  __syncthreads();
  for (int s = 32; s > 0; s >>= 1) {
    if (c < s) { s1[c] += s1[c + s]; s2[c] += s2[c + s]; }
    __syncthreads();
  }
  float mu  = s1[0] * (1.0f / C_);
  float var = s2[0] * (1.0f / C_) - mu * mu;
  float nrm = (val - mu) * rsqrtf(var + 1e-5f);
  h[(size_t)np * C_ + c] = nrm * gamma[c] + beta[c];
}

// ---------------- MLP1: [NO,64]@[64,256] + bias -> exact GELU ----------------
__global__ void k_mlp1(const float* __restrict__ h, const float* __restrict__ lin1_w,
                       const float* __restrict__ lin1_b, float* __restrict__ h1) {
  const int MT = NO_ / 16;   // 7500; 4 col-groups of 64 over F_=256
  int gt = blockIdx.x * blockDim.x + threadIdx.x;
  int wave = gt >> 5, lane = gt & 31;
  if (wave >= MT * 4) return;
  int mt = wave >> 2, ng = wave & 3;
  int m = lane & 15, half = lane >> 4;
  int row0 = mt * 16;
  const float* ap = h + (size_t)(row0 + m) * C_ + half * 2;
  v8f acc[4] = {};
#pragma unroll 4
  for (int k0 = 0; k0 < C_; k0 += 4) {
    v2f a = ld2(ap + k0);
#pragma unroll
    for (int t = 0; t < 4; ++t) {
      int col = ng * 64 + t * 16 + m;                 // B[c][f] = lin1_w[f*64+c]
      v2f b = ld2(lin1_w + (size_t)col * C_ + half * 2 + k0);
      acc[t] = wmma4(a, b, acc[t]);
    }
  }
#pragma unroll
  for (int t = 0; t < 4; ++t) {
    int col = ng * 64 + t * 16 + m;
    float bias = lin1_b[col];
#pragma unroll
    for (int v = 0; v < 8; ++v) {
      int row = row0 + v + 8 * half;
      float u = acc[t][v] + bias;
      float g = 0.5f * u * (1.0f + erff(u * 0.7071067811865475f));  // exact GELU
      h1[(size_t)row * F_ + col] = g;
    }
  }
}

// ---------------- MLP2: [NO,256]@[256,64] + bias; out = x + ls*h2 ----------------
__global__ void k_mlp2(const float* __restrict__ h1, const float* __restrict__ lin2_w,
                       const float* __restrict__ lin2_b, const float* __restrict__ x,
                       const float* __restrict__ lscale, float* __restrict__ out) {
  const int MT = NO_ / 16;   // 7500; one wave covers all 4 col tiles (C_=64)
  int gt = blockIdx.x * blockDim.x + threadIdx.x;
  int wave = gt >> 5, lane = gt & 31;
  if (wave >= MT) return;
  int m = lane & 15, half = lane >> 4;
  int row0 = wave * 16;
  const float* ap = h1 + (size_t)(row0 + m) * F_ + half * 2;
  v8f acc[4] = {};
  for (int k0 = 0; k0 < F_; k0 += 4) {
    v2f a = ld2(ap + k0);
#pragma unroll
    for (int t = 0; t < 4; ++t) {
      int col = t * 16 + m;                           // B[f][c] = lin2_w[c*256+f]
      v2f b = ld2(lin2_w + (size_t)col * F_ + half * 2 + k0);
      acc[t] = wmma4(a, b, acc[t]);
    }
  }
#pragma unroll
  for (int t = 0; t < 4; ++t) {
    int col = t * 16 + m;
    float bias = lin2_b[col];
    float ls   = lscale[col];
#pragma unroll
    for (int v = 0; v < 8; ++v) {
      size_t idx = (size_t)(row0 + v + 8 * half) * C_ + col;
      out[idx] = x[idx] + ls * (acc[t][v] + bias);
    }
  }
}

extern "C" void kernel_launch(void* const* d_in, const int* in_sizes, int n_in,
                              void* d_out, int out_size, void* d_ws, size_t ws_size,
                              hipStream_t stream) {
  const float* x        = (const float*)d_in[0];
  const float* kb       = (const float*)d_in[1];
  const float* fkb      = (const float*)d_in[2];
  const int*   ei       = (const int*)d_in[3];
  const int*   src      = ei;
  const int*   dst      = ei + E_;
  const float* kernel_w = (const float*)d_in[4];
  const float* fiber_w  = (const float*)d_in[5];
  const float* key_w    = (const float*)d_in[6];
  const float* key_b    = (const float*)d_in[7];
  const float* query_w  = (const float*)d_in[8];
  const float* query_b  = (const float*)d_in[9];
  const float* conv_b   = (const float*)d_in[10];
  const float* ln_g     = (const float*)d_in[11];
  const float* ln_b     = (const float*)d_in[12];
  const float* lin1_w   = (const float*)d_in[13];
  const float* lin1_b   = (const float*)d_in[14];
  const float* lin2_w   = (const float*)d_in[15];
  const float* lin2_b   = (const float*)d_in[16];
  const float* lscale   = (const float*)d_in[17];
  float* out = (float*)d_out;

  // workspace layout (16B-aligned offsets for float2/float4 loads)
  float* ws = (float*)d_ws;
  size_t off = 0;
  auto alloc = [&](size_t nfloats) {
    float* p = ws + off;
    off += (nfloats + 3) & ~(size_t)3;
    return p;
  };
  float* keys  = alloc((size_t)NO_ * KEY_);
  float* qrys  = alloc((size_t)NO_ * KEY_);
  float* ev    = alloc((size_t)EO_);        // logits, then exp values (in place)
  float* denom = alloc((size_t)NO_);
  float* gmax  = alloc(4);
  float* x1    = alloc((size_t)NO_ * C_);
  float* h     = alloc((size_t)NO_ * C_);
  float* h1    = alloc((size_t)NO_ * F_);
  float* fk    = alloc((size_t)O_ * O_ * C_);
  (void)ws_size; (void)in_sizes; (void)n_in; (void)out_size;

  k_init<<<(NO_ * C_ + 255) / 256, 256, 0, stream>>>(denom, x1, gmax);
  k_fk<<<(O_ * O_ * C_ + 255) / 256, 256, 0, stream>>>(fkb, fiber_w, fk);
  {
    int waves = (NO_ / 16) * (KEY_ / 16) * 2;
    k_qk<<<(waves * 32 + 255) / 256, 256, 0, stream>>>(x, key_w, key_b, query_w,
                                                       query_b, keys, qrys);
  }
  k_logits<<<(EO_ + 255) / 256, 256, 0, stream>>>(keys, qrys, src, dst, ev, gmax);
  k_expdenom<<<(EO_ + 255) / 256, 256, 0, stream>>>(ev, dst, gmax, denom);
  {
    int waves = EO_ / 16;
    k_edge<<<(waves * 32 + 255) / 256, 256, 0, stream>>>(kb, kernel_w, dst, ev,
                                                         denom, x1);
  }
  k_fiberln<<<NO_, 64, 0, stream>>>(x1, fk, conv_b, ln_g, ln_b, h);
  {
    int waves = (NO_ / 16) * 4;
    k_mlp1<<<(waves * 32 + 255) / 256, 256, 0, stream>>>(h, lin1_w, lin1_b, h1);
  }
  {
    int waves = NO_ / 16;
    k_mlp2<<<(waves * 32 + 255) / 256, 256, 0, stream>>>(h1, lin2_w, lin2_b, x,
                                                         lscale, out);
  }
}